// YOLOLoss_53601191854228
// MI455X (gfx1250) — compile-verified
//
#include <hip/hip_runtime.h>
#include <hip/hip_bf16.h>

// ---------------------------------------------------------------------------
// Fused YOLO loss for MI455X (gfx1250, wave32).
//
// Stage 1: one wave per (scale, b, t) target -> 3*8*32 = 768 waves.
//   - gather 85 channels at the assigned cell (≈261 KB total traffic)
//   - lanes 0..31 own class channels {l, l+32, l+64}
//   - all lanes redundantly compute box (DIoU) + obj losses (broadcast loads)
//   - 32-lane class-BCE reduction done on the matrix pipe:
//       V_WMMA_F32_16X16X4_F32 with A = lane partials (16x4, K1/K3 = 0),
//       B = ones  =>  D[m][*] = contrib_m + contrib_{m+16};
//       8 adds + shfl_xor(16) complete the reduction.
//   - per-target loss written to workspace (deterministic, no atomics)
// Stage 2: one wave32 workgroup reduces the 768 partials in fixed order,
//   closes with the same WMMA reduction, lane 0 writes d_out[0].
// ---------------------------------------------------------------------------

typedef __attribute__((ext_vector_type(2))) float v2f;
typedef __attribute__((ext_vector_type(8))) float v8f;

#define NUM_CLASSES 80
#define YEPS 1e-7f

__device__ __constant__ float c_ANC[3][3][2] = {
    {{10.f, 13.f}, {16.f, 30.f}, {33.f, 23.f}},
    {{30.f, 61.f}, {62.f, 45.f}, {59.f, 119.f}},
    {{116.f, 90.f}, {156.f, 198.f}, {373.f, 326.f}}};

__device__ __forceinline__ float sigmoidf_(float x) {
  return 1.0f / (1.0f + expf(-x));
}

// Reduce one float per lane across the wave using the matrix pipe.
// Returns the 32-lane sum in every lane.
__device__ __forceinline__ float wave32_sum_wmma(float contrib) {
  // A is a 16x4 f32 matrix: lanes 0-15 hold (K0,K1), lanes 16-31 hold (K2,K3)
  // for row M = lane % 16. Put the payload in K0/K2, zero K1/K3.
  v2f A;
  A.x = contrib;
  A.y = 0.0f;
  v2f B;            // all-ones 4x16 B => D[m][n] = rowsum(A[m][:])
  B.x = 1.0f;
  B.y = 1.0f;
  v8f C = {};
  v8f D = __builtin_amdgcn_wmma_f32_16x16x4_f32(
      /*neg_a=*/false, A, /*neg_b=*/false, B,
      /*c_mod=*/(short)0, C, /*reuse_a=*/false, /*reuse_b=*/false);
  // D VGPR j: lanes 0-15 -> row j, lanes 16-31 -> row 8+j (value uniform in n).
  float s8 = D[0] + D[1] + D[2] + D[3] + D[4] + D[5] + D[6] + D[7];
  // lanes 0-15 hold sum of rows 0-7, lanes 16-31 hold rows 8-15: combine.
  return s8 + __shfl_xor(s8, 16, 32);
}

__global__ void __launch_bounds__(256)
yolo_gather_kernel(const float* __restrict__ p0, const float* __restrict__ p1,
                   const float* __restrict__ p2,
                   const float* __restrict__ boxes,
                   const int* __restrict__ labels,
                   float* __restrict__ partial) {
  const int lane = threadIdx.x & 31;
  const int wave = (blockIdx.x * blockDim.x + threadIdx.x) >> 5;  // [0,768)

  const int scale = wave >> 8;   // /256
  const int rem = wave & 255;
  const int b = rem >> 5;        // /32
  const int t = rem & 31;

  const int nx = 80 >> scale;    // 80,40,20 (ny == nx)
  const float strideF = (float)(8 << scale);
  const float* __restrict__ p = (scale == 0) ? p0 : ((scale == 1) ? p1 : p2);

  // --- target box (wave-uniform; all lanes load same address) -------------
  const float4 bx = *(const float4*)(boxes + (size_t)(b * 32 + t) * 4);
  const float nxf = (float)nx;
  const float gxc = bx.x * nxf, gyc = bx.y * nxf;  // faithful: y also * nx
  const float gw = bx.z * nxf, gh = bx.w * nxf;

  int gi = (int)floorf(gxc);
  gi = (gi < 0) ? 0 : ((gi > nx - 1) ? nx - 1 : gi);
  int gj = (int)floorf(gyc);
  gj = (gj < 0) ? 0 : ((gj > nx - 1) ? nx - 1 : gj);

  // --- anchor assignment: argmin over anchors of max wh ratio -------------
  int a = 0;
  float best = 3.4e38f;
  for (int k = 0; k < 3; ++k) {
    const float aw = c_ANC[scale][k][0] / strideF;
    const float ah = c_ANC[scale][k][1] / strideF;
    float rw = gw / aw;
    rw = fmaxf(rw, 1.0f / rw);
    float rh = gh / ah;
    rh = fmaxf(rh, 1.0f / rh);
    const float m = fmaxf(rw, rh);
    if (m < best) { best = m; a = k; }   // strict <  -> first anchor on tie
  }

  // --- gather base: p[b, a*85 + ch, gj, gi], channel stride = nx*nx -------
  const size_t chs = (size_t)nx * (size_t)nx;
  const float* __restrict__ pb =
      p + ((size_t)b * 255 + (size_t)a * 85) * chs + (size_t)gj * nx + gi;

  // --- class BCE partial: lane l owns classes l, l+32, l+64 ---------------
  const int lbl = labels[b * 32 + t];
  float cpart = 0.0f;
#pragma unroll
  for (int g = 0; g < 3; ++g) {
    const int c = lane + g * 32;
    if (c < NUM_CLASSES) {
      const float x = pb[(size_t)(5 + c) * chs];
      const float s = sigmoidf_(x);
      const float y = (c == lbl) ? 1.0f : 0.0f;
      // bce_with_logits applied to sigmoided value s in (0,1):
      //   max(s,0) - s*y + log1p(exp(-|s|)) = s*(1-y) + log1p(exp(-s))
      cpart += s * (1.0f - y) + log1pf(expf(-s));
    }
  }

  // --- box + obj losses (all lanes redundantly; broadcast loads) ----------
  const float x0 = pb[0];
  const float x1 = pb[chs];
  const float x2 = pb[2 * chs];
  const float x3 = pb[3 * chs];
  const float x4 = pb[4 * chs];

  const float pcx = sigmoidf_(x0) + (float)gi;
  const float pcy = sigmoidf_(x1) + (float)gj;
  const float pw = expf(x2) * c_ANC[scale][a][0] / strideF;
  const float ph = expf(x3) * c_ANC[scale][a][1] / strideF;
  const float tx = gxc - (float)gi;
  const float ty = gyc - (float)gj;

  // DIoU (xywh), EPS placement faithful to reference (h1/h2 only)
  const float b1x1 = pcx - pw * 0.5f, b1x2 = pcx + pw * 0.5f;
  const float b1y1 = pcy - ph * 0.5f, b1y2 = pcy + ph * 0.5f;
  const float b2x1 = tx - gw * 0.5f, b2x2 = tx + gw * 0.5f;
  const float b2y1 = ty - gh * 0.5f, b2y2 = ty + gh * 0.5f;
  const float iw = fmaxf(fminf(b1x2, b2x2) - fmaxf(b1x1, b2x1), 0.0f);
  const float ih = fmaxf(fminf(b1y2, b2y2) - fmaxf(b1y1, b2y1), 0.0f);
  const float inter = iw * ih;
  const float w1 = b1x2 - b1x1, h1 = b1y2 - b1y1 + YEPS;
  const float w2 = b2x2 - b2x1, h2 = b2y2 - b2y1 + YEPS;
  const float uni = w1 * h1 + w2 * h2 - inter + YEPS;
  const float iou = inter / uni;
  const float cw = fmaxf(b1x2, b2x2) - fminf(b1x1, b2x1);
  const float chh = fmaxf(b1y2, b2y2) - fminf(b1y1, b2y1);
  const float c2 = cw * cw + chh * chh + YEPS;
  const float dx = b2x1 + b2x2 - b1x1 - b1x2;
  const float dy = b2y1 + b2y2 - b1y1 - b1y2;
  const float rho2 = (dx * dx + dy * dy) * 0.25f;
  const float diou = iou - rho2 / c2;

  const float loss_box = (1.0f - diou) * 5.0f;                 // LAMBDA_COORD
  const float sobj = sigmoidf_(x4);
  const float loss_obj = log1pf(expf(-sobj));                  // bce(s, 1)

  // --- wave reduction of class partials on the matrix pipe ----------------
  const float cls_sum = wave32_sum_wmma(cpart);

  if (lane == 0) {
    partial[wave] = cls_sum * (1.0f / (float)NUM_CLASSES) + loss_box + loss_obj;
  }
}

__global__ void __launch_bounds__(32)
yolo_reduce_kernel(const float* __restrict__ partial, float* __restrict__ out) {
  const int lane = threadIdx.x;  // single wave32 workgroup
  float acc = 0.0f;
  // fixed per-lane ascending order -> deterministic association
  for (int i = lane; i < 768; i += 32) acc += partial[i];
  const float total = wave32_sum_wmma(acc);
  if (lane == 0) out[0] = total;
}

extern "C" void kernel_launch(void* const* d_in, const int* in_sizes, int n_in,
                              void* d_out, int out_size, void* d_ws,
                              size_t ws_size, hipStream_t stream) {
  (void)in_sizes; (void)n_in; (void)out_size; (void)ws_size;
  const float* p0 = (const float*)d_in[0];
  const float* p1 = (const float*)d_in[1];
  const float* p2 = (const float*)d_in[2];
  const float* boxes = (const float*)d_in[3];
  const int* labels = (const int*)d_in[4];
  float* partial = (float*)d_ws;   // 768 floats, fully overwritten each call

  // 768 waves = 96 blocks x 256 threads (8 waves/block), all lanes active
  yolo_gather_kernel<<<96, 256, 0, stream>>>(p0, p1, p2, boxes, labels,
                                             partial);
  yolo_reduce_kernel<<<1, 32, 0, stream>>>(partial, (float*)d_out);
}